// DIN_44006234915568
// MI455X (gfx1250) — compile-verified
//
#include <hip/hip_runtime.h>
#include <math.h>

typedef __attribute__((ext_vector_type(2))) float v2f;
typedef __attribute__((ext_vector_type(8))) float v8f;

#define T_   200
#define TP   224     // T padded to 7 chunks of 32
#define CH   32      // rows per chunk
#define NTHR 160     // 5 waves

__device__ __forceinline__ float sigmoidf_(float x) { return 1.0f / (1.0f + __expf(-x)); }

__global__ __launch_bounds__(NTHR) void din_kernel(
    const int* __restrict__ item, const int* __restrict__ hist, const int* __restrict__ sl,
    const int* __restrict__ cate_list,
    const float* __restrict__ item_table, const float* __restrict__ cate_table,
    const float* __restrict__ dense_w, const float* __restrict__ dense_b,
    const float* __restrict__ bn1_g, const float* __restrict__ bn1_b,
    const float* __restrict__ bn2_g, const float* __restrict__ bn2_b,
    const float* __restrict__ att_w1, const float* __restrict__ att_b1,
    const float* __restrict__ att_w2, const float* __restrict__ att_b2,
    const float* __restrict__ att_w3, const float* __restrict__ att_b3,
    const float* __restrict__ fc1_w, const float* __restrict__ fc1_b, const float* __restrict__ alpha1,
    const float* __restrict__ fc2_w, const float* __restrict__ fc2_b, const float* __restrict__ alpha2,
    const float* __restrict__ fc3_w, const float* __restrict__ fc3_b,
    float* __restrict__ out)
{
    __shared__ __align__(16) v2f   heS[CH][66];    // he rows: 64 data float2 + 2 pad (conflict-free)
    __shared__ __align__(16) float s1S[CH][84];    // GEMM1 output (pad 84 -> conflict-free A loads)
    __shared__ __align__(16) float s2S[CH][48];    // GEMM2 output (N padded to 48)
    __shared__ __align__(16) v2f   w2S2[40][48];   // att_w2 pre-paired rows, zero-padded cols
    __shared__ float ieS[128];
    __shared__ float biasS[80];
    __shared__ float scoreS[TP];
    __shared__ float wS[TP];
    __shared__ float red[NTHR];
    __shared__ float attS[128];
    __shared__ float xS[192];
    __shared__ float f1S[80];
    __shared__ float f2S[40];
    __shared__ float sred[2];

    const int b    = blockIdx.x;
    const int tid  = threadIdx.x;
    const int lane = tid & 31;
    const int wv   = tid >> 5;       // 0..4 (one wave per GEMM1 N-tile)
    const int half = lane >> 4;      // 0|1
    const int l16  = lane & 15;
    const int slb  = sl[b];

    const float RSQ    = 0.08838834764831845f;    // 1/sqrt(2H)
    const float PADV   = -4294967295.0f;          // -2^32 + 1
    const float BN_RSQ = 0.9995003746877732f;     // 1/sqrt(1 + 1e-3)

    // ---- phase 0: target-item embedding + stage att_w2 ----
    if (tid < 128) {
        int it = item[b];
        ieS[tid] = (tid < 64) ? item_table[it * 64 + tid]
                              : cate_table[cate_list[it] * 64 + (tid - 64)];
    }
    for (int i = tid; i < 40 * 48; i += NTHR) {
        int p = i / 48, c2 = i - p * 48;
        v2f v;
        if (c2 < 40) { v.x = att_w2[(2 * p) * 40 + c2]; v.y = att_w2[(2 * p + 1) * 40 + c2]; }
        else         { v.x = 0.0f; v.y = 0.0f; }
        w2S2[p][c2] = v;
    }
    __syncthreads();

    // per-b attention bias: b1 + q @ (W1q + W1d)
    if (tid < 80) {
        float acc = att_b1[tid];
        for (int i = 0; i < 128; ++i)
            acc += ieS[i] * (att_w1[i * 80 + tid] + att_w1[(256 + i) * 80 + tid]);
        biasS[tid] = acc;
    }

    // per-b folded weight W_b = (W1k - W1d) + diag(q) * W1m, kept in registers as
    // WMMA B-fragments for this wave's N-tile (K=128 -> 32 k-steps, float2/lane)
    v2f bfrag[32];
    {
        const int j = wv * 16 + l16;
#pragma unroll
        for (int kk = 0; kk < 32; ++kk) {
            int i0 = 4 * kk + 2 * half;
            float q0 = ieS[i0], q1 = ieS[i0 + 1];
            bfrag[kk].x = att_w1[(128 + i0) * 80 + j] - att_w1[(256 + i0) * 80 + j]
                        + q0 * att_w1[(384 + i0) * 80 + j];
            bfrag[kk].y = att_w1[(128 + i0 + 1) * 80 + j] - att_w1[(256 + i0 + 1) * 80 + j]
                        + q1 * att_w1[(384 + i0 + 1) * 80 + j];
        }
    }

    // ---- phase 1: attention scores, chunked over T; skip fully-masked chunks ----
    const int nact = (slb + CH - 1) / CH;
    for (int c = 0; c < nact; ++c) {
        // gather he chunk (float4 per thread-iteration)
        for (int i = tid; i < CH * 32; i += NTHR) {
            int row = i >> 5, d0 = (i & 31) * 4, t = c * CH + row;
            float4 v;
            if (t < T_) {
                int it = hist[b * T_ + t];
                v = (d0 < 64) ? *(const float4*)(item_table + it * 64 + d0)
                              : *(const float4*)(cate_table + cate_list[it] * 64 + (d0 - 64));
            } else {
                v = make_float4(0.f, 0.f, 0.f, 0.f);
            }
            float* hp = (float*)(&heS[row][0]);
            *(float4*)(hp + d0) = v;
        }
        __syncthreads();

        // GEMM1: [32 x 128] @ W_b[128 x 80]; wave wv owns N-tile wv; loop 2 M-tiles
        for (int mt = 0; mt < 2; ++mt) {
            v8f acc = {0.f, 0.f, 0.f, 0.f, 0.f, 0.f, 0.f, 0.f};
            const int arow = mt * 16 + l16;
#pragma unroll
            for (int kk = 0; kk < 32; ++kk) {
                v2f a = heS[arow][2 * kk + half];
                acc = __builtin_amdgcn_wmma_f32_16x16x4_f32(
                          false, a, false, bfrag[kk], (short)0, acc, false, false);
            }
            const int col   = wv * 16 + l16;
            const int rbase = mt * 16 + half * 8;
            const float bias = biasS[col];
#pragma unroll
            for (int r = 0; r < 8; ++r)
                s1S[rbase + r][col] = sigmoidf_(acc[r] + bias);
        }
        __syncthreads();

        // GEMM2: [32 x 80] @ att_w2[80 x 40(pad 48)]; 6 tile-jobs over 5 waves
        for (int job = wv; job < 6; job += 5) {
            int mt = job / 3, nt = job - mt * 3;
            v8f acc = {0.f, 0.f, 0.f, 0.f, 0.f, 0.f, 0.f, 0.f};
            const int arow = mt * 16 + l16;
            const int col  = nt * 16 + l16;
#pragma unroll
            for (int kk = 0; kk < 20; ++kk) {
                v2f a  = *(const v2f*)&s1S[arow][4 * kk + 2 * half];
                v2f bb = w2S2[2 * kk + half][col];
                acc = __builtin_amdgcn_wmma_f32_16x16x4_f32(
                          false, a, false, bb, (short)0, acc, false, false);
            }
            const int rbase = mt * 16 + half * 8;
#pragma unroll
            for (int r = 0; r < 8; ++r) {
                float v = (col < 40) ? sigmoidf_(acc[r] + att_b2[col]) : 0.0f;
                s2S[rbase + r][col] = v;
            }
        }
        __syncthreads();

        // GEMM3: [32 x 40] @ w3[40 x 1]
        if (tid < CH) {
            float acc = att_b3[0];
            for (int j = 0; j < 40; ++j) acc += s2S[tid][j] * att_w3[j];
            scoreS[c * CH + tid] = acc;
        }
        __syncthreads();
    }

    // ---- masked softmax over T ----
    float lmax = -3.4e38f;
    for (int t = tid; t < TP; t += NTHR) {
        float v = ((t < slb) ? scoreS[t] : PADV) * RSQ;
        scoreS[t] = v;
        lmax = fmaxf(lmax, v);
    }
    red[tid] = lmax;
    __syncthreads();
    if (tid < 32) {
        float m = red[tid];
#pragma unroll
        for (int k = 1; k < 5; ++k) m = fmaxf(m, red[tid + 32 * k]);
#pragma unroll
        for (int off = 16; off; off >>= 1) m = fmaxf(m, __shfl_xor(m, off));
        if (tid == 0) sred[0] = m;
    }
    __syncthreads();
    const float M = sred[0];
    float lsum = 0.f;
    for (int t = tid; t < TP; t += NTHR) {
        float e = __expf(scoreS[t] - M);
        wS[t] = e;
        lsum += e;
    }
    red[tid] = lsum;
    __syncthreads();
    if (tid < 32) {
        float s = red[tid];
#pragma unroll
        for (int k = 1; k < 5; ++k) s += red[tid + 32 * k];
#pragma unroll
        for (int off = 16; off; off >>= 1) s += __shfl_xor(s, off);
        if (tid == 0) sred[1] = s;
    }
    __syncthreads();
    const float invS = 1.0f / sred[1];
    for (int t = tid; t < TP; t += NTHR) wS[t] *= invS;
    __syncthreads();

    // ---- weighted sum of he (re-gather, L2-resident) + BN1 ----
    if (tid < 128) {
        const int d = tid;
        float acc = 0.f;
        for (int t = 0; t < slb; ++t) {
            int it = hist[b * T_ + t];
            float v = (d < 64) ? item_table[it * 64 + d]
                               : cate_table[cate_list[it] * 64 + (d - 64)];
            acc += wS[t] * v;
        }
        attS[d] = acc * (bn1_g[d] * BN_RSQ) + bn1_b[d];
    }
    __syncthreads();

    // ---- dense (128->64), concat with ie, BN2 ----
    if (tid < 64) {
        float acc = dense_b[tid];
        for (int d = 0; d < 128; ++d) acc += attS[d] * dense_w[d * 64 + tid];
        xS[tid] = acc;
    }
    for (int k = tid; k < 128; k += NTHR) xS[64 + k] = ieS[k];
    __syncthreads();
    for (int c2 = tid; c2 < 192; c2 += NTHR)
        xS[c2] = xS[c2] * (bn2_g[c2] * BN_RSQ) + bn2_b[c2];
    __syncthreads();

    // ---- FC head ----
    if (tid < 80) {
        float acc = fc1_b[tid];
        for (int c2 = 0; c2 < 192; ++c2) acc += xS[c2] * fc1_w[c2 * 80 + tid];
        float s = sigmoidf_(acc);
        f1S[tid] = (s >= 0.f) ? s : alpha1[tid] * s;
    }
    __syncthreads();
    if (tid < 40) {
        float acc = fc2_b[tid];
        for (int c2 = 0; c2 < 80; ++c2) acc += f1S[c2] * fc2_w[c2 * 40 + tid];
        float s = sigmoidf_(acc);
        f2S[tid] = (s >= 0.f) ? s : alpha2[tid] * s;
    }
    __syncthreads();
    if (tid == 0) {
        float acc = fc3_b[0];
        for (int j = 0; j < 40; ++j) acc += f2S[j] * fc3_w[j];
        out[b] = sigmoidf_(acc);
    }
}

extern "C" void kernel_launch(void* const* d_in, const int* in_sizes, int n_in,
                              void* d_out, int out_size, void* d_ws, size_t ws_size,
                              hipStream_t stream) {
    (void)n_in; (void)out_size; (void)d_ws; (void)ws_size;
    const int*   item       = (const int*)d_in[0];
    const int*   hist       = (const int*)d_in[1];
    const int*   sl         = (const int*)d_in[2];
    const int*   cate_list  = (const int*)d_in[3];
    const float* item_table = (const float*)d_in[4];
    const float* cate_table = (const float*)d_in[5];
    const float* dense_w    = (const float*)d_in[6];
    const float* dense_b    = (const float*)d_in[7];
    const float* bn1_g      = (const float*)d_in[8];
    const float* bn1_b      = (const float*)d_in[9];
    const float* bn2_g      = (const float*)d_in[10];
    const float* bn2_b      = (const float*)d_in[11];
    const float* att_w1     = (const float*)d_in[12];
    const float* att_b1     = (const float*)d_in[13];
    const float* att_w2     = (const float*)d_in[14];
    const float* att_b2     = (const float*)d_in[15];
    const float* att_w3     = (const float*)d_in[16];
    const float* att_b3     = (const float*)d_in[17];
    const float* fc1_w      = (const float*)d_in[18];
    const float* fc1_b      = (const float*)d_in[19];
    const float* alpha1     = (const float*)d_in[20];
    const float* fc2_w      = (const float*)d_in[21];
    const float* fc2_b      = (const float*)d_in[22];
    const float* alpha2     = (const float*)d_in[23];
    const float* fc3_w      = (const float*)d_in[24];
    const float* fc3_b      = (const float*)d_in[25];

    const int B = in_sizes[0];   // 1024
    din_kernel<<<dim3(B), dim3(NTHR), 0, stream>>>(
        item, hist, sl, cate_list, item_table, cate_table,
        dense_w, dense_b, bn1_g, bn1_b, bn2_g, bn2_b,
        att_w1, att_b1, att_w2, att_b2, att_w3, att_b3,
        fc1_w, fc1_b, alpha1, fc2_w, fc2_b, alpha2, fc3_w, fc3_b,
        (float*)d_out);
}